// MCTNE_73160472920287
// MI455X (gfx1250) — compile-verified
//
#include <hip/hip_runtime.h>

#define NODES 100000
#define EMB 128
#define B_SZ 4096
#define H_SZ 50
#define K_SZ 10
#define NTYPE 24
#define NWAVE 4

typedef float v2f __attribute__((ext_vector_type(2)));
typedef float v8f __attribute__((ext_vector_type(8)));
typedef int   v4i __attribute__((ext_vector_type(4)));
typedef __attribute__((address_space(1))) v4i gv4i;  // global int4
typedef __attribute__((address_space(3))) v4i lv4i;  // LDS int4

#if defined(__AMDGCN__) && __has_builtin(__builtin_amdgcn_global_load_async_to_lds_b128)
#define HAVE_ASYNC 1
#else
#define HAVE_ASYNC 0
#endif

__device__ __forceinline__ float wave_sum(float v) {
  #pragma unroll
  for (int off = 16; off > 0; off >>= 1) v += __shfl_xor(v, off, 32);
  return v;
}
__device__ __forceinline__ float wave_max(float v) {
  #pragma unroll
  for (int off = 16; off > 0; off >>= 1) v = fmaxf(v, __shfl_xor(v, off, 32));
  return v;
}

// Stage one 512B embedding row (32 lanes x 16B) global -> LDS.
__device__ __forceinline__ void stage_row(const float* gsrc, float* ldst, int lane) {
#if HAVE_ASYNC
  __builtin_amdgcn_global_load_async_to_lds_b128(
      (gv4i*)(gsrc + lane * 4), (lv4i*)(ldst + lane * 4), 0, 0);
#else
  float4 v = *(const float4*)(gsrc + lane * 4);
  *(float4*)(ldst + lane * 4) = v;
#endif
}

__device__ __forceinline__ void wait_stage() {
#if HAVE_ASYNC
 #if __has_builtin(__builtin_amdgcn_s_wait_asynccnt)
  __builtin_amdgcn_s_wait_asynccnt(0);
 #else
  asm volatile("s_wait_asynccnt 0x0" ::: "memory");
 #endif
#endif
}

__global__ __launch_bounds__(32 * NWAVE)
void mctne_kernel(const int* __restrict__ idx_p,
                  const int* __restrict__ s_nodes, const int* __restrict__ t_nodes,
                  const float* __restrict__ t_times,
                  const int* __restrict__ n_nodes, const int* __restrict__ h_nodes,
                  const float* __restrict__ h_times, const int* __restrict__ h_types,
                  const float* __restrict__ h_t_time,
                  const float* __restrict__ h_time_mask, const float* __restrict__ h_t_masks,
                  const float* __restrict__ node_emb, const float* __restrict__ delta,
                  const float* __restrict__ att_param,
                  float* __restrict__ out)
{
  __shared__ float sh_h[64 * EMB];     // h rows, padded to 64 (rows >=50 zeroed)
  __shared__ float sh_n[16 * EMB];     // n rows, padded to 16 (rows >=10 zeroed)
  __shared__ float sh_s[EMB];
  __shared__ float sh_t[EMB];
  __shared__ float sh_hn[64 * 16];     // hn[j][k] from WMMA
  __shared__ float sh_hh[H_SZ];        // ||h_j||^2
  __shared__ float sh_pa[H_SZ];        // -||h_j - t||^2
  __shared__ float sh_pta[H_SZ];       // -||h_j - s||^2
  __shared__ float sh_wj[H_SZ];        // att*e_dt*h_time_mask
  __shared__ float sh_aet[H_SZ];       // att*e_dtt
  __shared__ float sh_logit[H_SZ];
  __shared__ float sh_nn[K_SZ];        // ||n_k||^2
  __shared__ float sh_nmu[K_SZ];       // -||s - n_k||^2
  __shared__ float sh_nta[K_SZ];       // -||t - n_k||^2

  const int tid  = threadIdx.x;
  const int wave = tid >> 5;
  const int lane = tid & 31;
  const int b    = blockIdx.x;
  const int idx  = idx_p[0];
  const float* emb = node_emb + (size_t)idx * NODES * EMB;

  const int   sn  = s_nodes[b];
  const int   tn  = t_nodes[b];
  const float tt  = t_times[b];
  const float dlt = delta[sn];

  // ---- Phase 1: stage all 62 gathered rows into LDS (async path, all in flight) ----
  // row ids: 0 -> s, 1 -> t, 2..51 -> h[j], 52..61 -> n[k]
  for (int r = wave; r < 2 + H_SZ + K_SZ; r += NWAVE) {
    int node; float* dst;
    if (r == 0)      { node = sn;                           dst = sh_s; }
    else if (r == 1) { node = tn;                           dst = sh_t; }
    else if (r < 2 + H_SZ) { node = h_nodes[b * H_SZ + (r - 2)];   dst = &sh_h[(r - 2) * EMB]; }
    else             { node = n_nodes[b * K_SZ + (r - 52)]; dst = &sh_n[(r - 52) * EMB]; }
    stage_row(emb + (size_t)node * EMB, dst, lane);
  }
  // zero the pad rows (so WMMA operand loads need no masking and EXEC stays all-1s)
  for (int i = tid; i < (64 - H_SZ) * EMB; i += 32 * NWAVE) sh_h[H_SZ * EMB + i] = 0.f;
  for (int i = tid; i < (16 - K_SZ) * EMB; i += 32 * NWAVE) sh_n[K_SZ * EMB + i] = 0.f;
  wait_stage();
  __syncthreads();

  // ---- Phase 2: per-row stats from LDS (rows split across the 4 waves) ----
  for (int j = wave; j < H_SZ; j += NWAVE) {
    float4 v  = *(const float4*)&sh_h[j * EMB + lane * 4];
    float4 s4 = *(const float4*)&sh_s[lane * 4];
    float4 t4 = *(const float4*)&sh_t[lane * 4];
    float a0 = v.x*v.x + v.y*v.y + v.z*v.z + v.w*v.w;
    float dx = v.x - t4.x, dy = v.y - t4.y, dz = v.z - t4.z, dw = v.w - t4.w;
    float a1 = dx*dx + dy*dy + dz*dz + dw*dw;
    dx = v.x - s4.x; dy = v.y - s4.y; dz = v.z - s4.z; dw = v.w - s4.w;
    float a2 = dx*dx + dy*dy + dz*dz + dw*dw;
    a0 = wave_sum(a0); a1 = wave_sum(a1); a2 = wave_sum(a2);
    if (lane == 0) { sh_hh[j] = a0; sh_pa[j] = -a1; sh_pta[j] = -a2; }
  }
  for (int k = wave; k < K_SZ; k += NWAVE) {
    float4 v  = *(const float4*)&sh_n[k * EMB + lane * 4];
    float4 s4 = *(const float4*)&sh_s[lane * 4];
    float4 t4 = *(const float4*)&sh_t[lane * 4];
    float a0 = v.x*v.x + v.y*v.y + v.z*v.z + v.w*v.w;
    float dx = s4.x - v.x, dy = s4.y - v.y, dz = s4.z - v.z, dw = s4.w - v.w;
    float a1 = dx*dx + dy*dy + dz*dz + dw*dw;
    dx = t4.x - v.x; dy = t4.y - v.y; dz = t4.z - v.z; dw = t4.w - v.w;
    float a2 = dx*dx + dy*dy + dz*dz + dw*dw;
    a0 = wave_sum(a0); a1 = wave_sum(a1); a2 = wave_sum(a2);
    if (lane == 0) { sh_nn[k] = a0; sh_nmu[k] = -a1; sh_nta[k] = -a2; }
  }
  __syncthreads();

  // ---- Phase 3 (wave 0): p_mu, softmax(att), decay weights, p_lambda ----
  if (wave == 0) {
    float4 s4 = *(const float4*)&sh_s[lane * 4];
    float4 t4 = *(const float4*)&sh_t[lane * 4];
    float dx = s4.x - t4.x, dy = s4.y - t4.y, dz = s4.z - t4.z, dw = s4.w - t4.w;
    const float p_mu = -wave_sum(dx*dx + dy*dy + dz*dz + dw*dw);

    float m = -3.4e38f;
    for (int j = lane; j < H_SZ; j += 32) {
      float lg = att_param[(size_t)sn * NTYPE + h_types[b * H_SZ + j]];
      sh_logit[j] = lg;
      m = fmaxf(m, lg);
    }
    m = wave_max(m);
    float ssum = 0.f;
    for (int j = lane; j < H_SZ; j += 32) ssum += expf(sh_logit[j] - m);
    ssum = wave_sum(ssum);
    const float inv = 1.f / ssum;

    float psum = 0.f;
    for (int j = lane; j < H_SZ; j += 32) {
      float att  = expf(sh_logit[j] - m) * inv;
      float edt  = expf(dlt * fabsf(tt - h_times[b * H_SZ + j]));
      float edtt = expf(dlt * fabsf(tt - h_t_time[b * H_SZ + j]));
      float htm  = h_t_masks[b * H_SZ + j];
      float hm   = h_time_mask[b * H_SZ + j];
      psum += att * (sh_pa[j] * edt * htm + sh_pta[j] * edtt * hm);
      sh_wj[j]  = att * edt * hm;   // weight on n_alpha (reference uses h_time_mask here)
      sh_aet[j] = att * edtt;       // weight on n_t_alpha (no mask in reference)
    }
    psum = wave_sum(psum);
    if (lane == 0) out[b] = p_mu + psum;
  }
  __syncthreads();

  // ---- Phase 4: WMMA hn[j][k] = h_j . n_k  (f32 16x16x4; wave w owns M-tile w) ----
  // A 16x4: lane<16 -> M=lane, regs K=0,1 ; lane>=16 -> M=lane-16, regs K=2,3
  // B 4x16: lane&15 = N ; lane<16 -> regs K=0,1 ; lane>=16 -> regs K=2,3
  {
    const int half = lane >> 4;
    const int l16  = lane & 15;
    const int row  = wave * 16 + l16;
    v8f acc = {0.f, 0.f, 0.f, 0.f, 0.f, 0.f, 0.f, 0.f};
    for (int e = 0; e < EMB; e += 4) {
      v2f a, bb;
      a.x  = sh_h[row * EMB + e + 2 * half];
      a.y  = sh_h[row * EMB + e + 2 * half + 1];
      bb.x = sh_n[l16 * EMB + e + 2 * half];
      bb.y = sh_n[l16 * EMB + e + 2 * half + 1];
      acc = __builtin_amdgcn_wmma_f32_16x16x4_f32(
          /*neg_a=*/false, a, /*neg_b=*/false, bb,
          /*c_mod=*/(short)0, acc, /*reuse_a=*/false, /*reuse_b=*/false);
    }
    // C/D layout: VGPR r -> M = wave*16 + r + 8*half, N = l16
    #pragma unroll
    for (int r = 0; r < 8; r++)
      sh_hn[(wave * 16 + r + 8 * half) * 16 + l16] = acc[r];
  }
  __syncthreads();

  // ---- Phase 5: n_lambda[k] = n_mu[k] + sum_j [ wj_j * -(hh_j + nn_k - 2 hn_jk) + aet_j * nta_k ]
  if (tid < K_SZ) {
    const float nnk = sh_nn[tid];
    const float nta = sh_nta[tid];
    float acc = 0.f;
    for (int j = 0; j < H_SZ; j++) {
      float na = -(sh_hh[j] + nnk - 2.f * sh_hn[j * 16 + tid]);
      acc += na * sh_wj[j] + nta * sh_aet[j];
    }
    out[B_SZ + b * K_SZ + tid] = sh_nmu[tid] + acc;
  }
}

extern "C" void kernel_launch(void* const* d_in, const int* in_sizes, int n_in,
                              void* d_out, int out_size, void* d_ws, size_t ws_size,
                              hipStream_t stream) {
  (void)in_sizes; (void)n_in; (void)out_size; (void)d_ws; (void)ws_size;
  // setup_inputs() dict order:
  const int*   idx         = (const int*)  d_in[0];
  const int*   s_nodes     = (const int*)  d_in[1];
  const int*   t_nodes     = (const int*)  d_in[2];
  const float* t_times     = (const float*)d_in[3];
  const int*   n_nodes     = (const int*)  d_in[4];
  const int*   h_nodes     = (const int*)  d_in[5];
  const float* h_times     = (const float*)d_in[6];
  const int*   h_types     = (const int*)  d_in[7];
  const float* h_t_time    = (const float*)d_in[8];
  const float* h_time_mask = (const float*)d_in[9];
  const float* h_t_masks   = (const float*)d_in[10];
  const float* node_emb    = (const float*)d_in[11];
  const float* delta       = (const float*)d_in[12];
  const float* att_param   = (const float*)d_in[13];
  // d_in[14] = att_param_p (unused by reference)
  float* out = (float*)d_out;

  mctne_kernel<<<dim3(B_SZ), dim3(32 * NWAVE), 0, stream>>>(
      idx, s_nodes, t_nodes, t_times, n_nodes, h_nodes, h_times, h_types,
      h_t_time, h_time_mask, h_t_masks, node_emb, delta, att_param, out);
}